// GatedBlock_17987323036062
// MI455X (gfx1250) — compile-verified
//
#include <hip/hip_runtime.h>
#include <hip/hip_bf16.h>

// GatedBlock for MI455X (gfx1250, wave32).
// y[b, q*128+n] = sum_p g[b,p,q] * x[b, p*128+m] * W[p*128+m, q*128+n] + bias
// Strategy: fold per-(row,p,q) gate scalar into A while staging to LDS, then
// dense f32 WMMA (V_WMMA_F32_16X16X4_F32) — exact f32 precision as reference.
// W tiles are staged LDS-side by the Tensor Data Mover (TENSOR_LOAD_TO_LDS),
// double-buffered and overlapped with compute via TENSORcnt.
// NOTE: TDM builtin called UNCONDITIONALLY this round to verify availability.

#define BATCH   8192
#define DIN     2048
#define DOUT    2048
#define BLK     128
#define S       16      // 2048/128
#define NB      256     // S*S gate outputs
#define KSEL    26      // ceil(0.1 * 256)

typedef float v2f __attribute__((ext_vector_type(2)));
typedef float v8f __attribute__((ext_vector_type(8)));

typedef unsigned int tdm_v4u __attribute__((ext_vector_type(4)));
typedef int          tdm_v4i __attribute__((ext_vector_type(4)));
typedef int          tdm_v8i __attribute__((ext_vector_type(8)));

__device__ __forceinline__ v8f wmma_f32_16x16x4(v2f a, v2f b, v8f c) {
    // 8 args: (neg_a, A, neg_b, B, c_mod, C, reuse_a, reuse_b)
    return __builtin_amdgcn_wmma_f32_16x16x4_f32(
        false, a, false, b, (short)0, c, false, false);
}

// Issue one TDM DMA: 2D tile, 32 rows x 128 f32 cols, row stride DOUT elems,
// from global `src` into LDS offset `lds_off`. LDS pad: 4 DWORDs after every
// 128 DWORDs (pad_interval=6 -> 128 dw, pad_amount=3 -> 4 dw) => LDS row
// stride 132 floats (also de-conflicts the two lane-half B reads).
__device__ __forceinline__ void tdm_load_w_slab(const float* src, unsigned lds_off) {
    unsigned long long ga = (unsigned long long)(uintptr_t)src;   // 57-bit VA
    tdm_v4u g0 = { 1u,                                            // count=1, user desc
                   lds_off,                                       // lds_addr [63:32]
                   (unsigned)(ga & 0xFFFFFFFFull),                // global_addr lo
                   (unsigned)((ga >> 32) & 0x01FFFFFFull)         // global_addr hi
                       | 0x80000000u };                           // type=2 ("image")
    tdm_v8i g1 = { (int)0x07920000u,   // data_size=4B | pad_en | interval=6 | amount=3
                   (int)(128u << 16),  // tensor_dim0 = 128  ([79:48] lo half)
                   (int)(32u  << 16),  // tensor_dim0 hi=0 | tensor_dim1 = 32
                   (int)(128u << 16),  // tensor_dim1 hi=0 | tile_dim0 = 128
                   32,                 // tile_dim1 = 32 | tile_dim2 = 0
                   (int)DOUT,          // tensor_dim0_stride = 2048
                   0, 0 };             // stride hi / tensor_dim1_stride = 0
    tdm_v4i gz = { 0, 0, 0, 0 };       // groups 2/3 unused (2D tensor)
#if defined(__clang_major__) && (__clang_major__ >= 23)
    tdm_v8i gz8 = { 0, 0, 0, 0, 0, 0, 0, 0 };
    __builtin_amdgcn_tensor_load_to_lds(g0, g1, gz, gz, gz8, 0);
#else
    __builtin_amdgcn_tensor_load_to_lds(g0, g1, gz, gz, 0);
#endif
}

// ---------------------------------------------------------------------------
// Kernel: gated GEMM. Workgroup = (q, 64-row batch tile): 64x128 C tile.
// 8 waves: wave w -> row group (w>>1)*16, col groups ((w&1)*4 + 0..3)*16.
// K swept in 64 slabs of 32 (p = slab>>2). Gate scalar folded into A at LDS
// staging; W slabs DMA'd into double-buffered LDS by the TDM (wave 0 issues,
// TENSORcnt-synchronized), overlapped with WMMA compute on the other buffer.
// ---------------------------------------------------------------------------
__global__ __launch_bounds__(256) void gated_gemm_kernel(
    const float* __restrict__ x, const float* __restrict__ w,
    const float* __restrict__ bias, const float* __restrict__ g,
    float* __restrict__ out)
{
    __shared__ float xs[64][36];             // 64 rows x 32 K (gate-scaled), pad->36
    __shared__ float ws[2][32 * 132];        // double buffer: 32 K x 128 cols, stride 132

    const int q    = blockIdx.x;             // 0..15 output block column
    const int row0 = blockIdx.y * 64;
    const int tid  = threadIdx.x;
    const int wave = tid >> 5;
    const int lane = tid & 31;
    const int ln   = lane & 15;
    const int kh   = lane >> 4;
    const int rg   = wave >> 1;              // row group 0..3
    const int cg0  = (wave & 1) * 4;         // first col group (of 4)
    const bool issuer = (wave == 0);         // uniform per wave

    // Escape ws so the compiler can't assume it is never stored (TDM writes it).
    asm volatile("" :: "v"(&ws[0][0]) : "memory");

    if (issuer) {
        tdm_load_w_slab(w + (size_t)q * BLK,                       // kb = 0 slab
                        (unsigned)(uintptr_t)(&ws[0][0]));
    }

    v8f acc[4] = {};

    for (int kb = 0; kb < 4 * S; ++kb) {
        const int p     = kb >> 2;
        const int kglob = p * BLK + (kb & 3) * 32;

        { // stage gate-scaled x: 64 rows x 32 K; 8 floats/thread
            int r  = tid >> 2;
            int c0 = (tid & 3) * 8;
            float gs = g[(size_t)(row0 + r) * NB + p * S + q];
            const float4* src = (const float4*)(x + (size_t)(row0 + r) * DIN + kglob + c0);
            float4* dst = (float4*)(&xs[r][c0]);
            #pragma unroll
            for (int i = 0; i < 2; ++i) {
                float4 v = src[i];
                v.x *= gs; v.y *= gs; v.z *= gs; v.w *= gs;
                dst[i] = v;
            }
            if (kb + 1 < 4 * S) {            // prefetch next x slab (global_prefetch_b8)
                const int nk = ((kb + 1) >> 2) * BLK + ((kb + 1) & 3) * 32;
                __builtin_prefetch(x + (size_t)(row0 + r) * DIN + nk + c0, 0, 1);
            }
        }

        if (issuer) {
            if (kb + 1 < 4 * S) {            // DMA next W slab into the other buffer
                const int nk = ((kb + 1) >> 2) * BLK + ((kb + 1) & 3) * 32;
                tdm_load_w_slab(w + (size_t)nk * DOUT + q * BLK,
                                (unsigned)(uintptr_t)(&ws[(kb + 1) & 1][0]));
                __builtin_amdgcn_s_wait_tensorcnt(1);  // in-order: older DMA done
            } else {
                __builtin_amdgcn_s_wait_tensorcnt(0);
            }
        }
        __syncthreads();

        const float* wsb = &ws[kb & 1][0];
        #pragma unroll
        for (int k = 0; k < 32; k += 4) {
            v2f a;
            a.x = xs[rg * 16 + ln][k + kh * 2];
            a.y = xs[rg * 16 + ln][k + kh * 2 + 1];
            #pragma unroll
            for (int t = 0; t < 4; ++t) {
                int col = (cg0 + t) * 16 + ln;
                v2f bf;
                bf.x = wsb[(k + kh * 2)     * 132 + col];
                bf.y = wsb[(k + kh * 2 + 1) * 132 + col];
                acc[t] = wmma_f32_16x16x4(a, bf, acc[t]);
            }
        }
        __syncthreads();
    }

    #pragma unroll
    for (int t = 0; t < 4; ++t) {
        int colg = q * BLK + (cg0 + t) * 16 + ln;
        float bv = bias[colg];
        #pragma unroll
        for (int r = 0; r < 8; ++r) {
            int row = row0 + rg * 16 + r + kh * 8;
            out[(size_t)row * DOUT + colg] = acc[t][r] + bv;
        }
    }
}

// ---------------------------------------------------------------------------
// Kernel: gate GEMM  graw[b, j] = x[b,:] @ gate_w[:, j] + gate_b[j]
//   M=8192, N=256, K=2048.  1 workgroup = 16 rows x 256 cols, 8 waves,
//   each wave owns two 16x16 C tiles. x chunk staged in LDS per 256-wide K slab.
// ---------------------------------------------------------------------------
__global__ __launch_bounds__(256) void gate_gemm_kernel(
    const float* __restrict__ x, const float* __restrict__ gw,
    const float* __restrict__ gb, float* __restrict__ graw)
{
    __shared__ float xs[16][260];            // stride 260 -> 16B-mult, banks +4/row

    const int row0 = blockIdx.x * 16;
    const int tid  = threadIdx.x;
    const int wave = tid >> 5;
    const int lane = tid & 31;
    const int ln   = lane & 15;              // M (A) / N (B,C) index
    const int kh   = lane >> 4;              // K-half selector

    v8f acc[2] = {};

    for (int kc = 0; kc < DIN; kc += 256) {
        { // stage 16 rows x 256 floats: 16 floats per thread
            int r  = tid >> 4;
            int c0 = (tid & 15) * 16;
            const float4* src = (const float4*)(x + (size_t)(row0 + r) * DIN + kc + c0);
            float4* dst = (float4*)(&xs[r][c0]);
            #pragma unroll
            for (int i = 0; i < 4; ++i) dst[i] = src[i];
        }
        __syncthreads();

        for (int k = 0; k < 256; k += 4) {
            v2f a;                            // A 16x4: lanes0-15 K{0,1}, lanes16-31 K{2,3}
            a.x = xs[ln][k + kh * 2];
            a.y = xs[ln][k + kh * 2 + 1];
            #pragma unroll
            for (int t = 0; t < 2; ++t) {
                int col = wave * 32 + t * 16 + ln;
                v2f b;                        // B 4x16: same K striping, N across lanes
                b.x = gw[(size_t)(kc + k + kh * 2)     * NB + col];
                b.y = gw[(size_t)(kc + k + kh * 2 + 1) * NB + col];
                acc[t] = wmma_f32_16x16x4(a, b, acc[t]);
            }
        }
        __syncthreads();
    }

    #pragma unroll
    for (int t = 0; t < 2; ++t) {
        int col = wave * 32 + t * 16 + ln;
        float bv = gb[col];
        #pragma unroll
        for (int r = 0; r < 8; ++r) {         // C: VGPR r -> M = r + kh*8
            graw[(size_t)(row0 + r + kh * 8) * NB + col] = acc[t][r] + bv;
        }
    }
}

// ---------------------------------------------------------------------------
// Kernel: Sparsify1D per row (in place). keep v iff count(strictly greater)<26
// (matches g >= topval incl. ties), then scale kept by 256/sum(kept).
// ---------------------------------------------------------------------------
__global__ __launch_bounds__(256) void sparsify_kernel(float* __restrict__ g)
{
    __shared__ float vals[NB];
    __shared__ float red[NB];

    const int b = blockIdx.x;
    const int t = threadIdx.x;
    float v = g[(size_t)b * NB + t];
    vals[t] = v;
    __syncthreads();

    int cnt = 0;
    #pragma unroll 8
    for (int j = 0; j < NB; ++j) cnt += (vals[j] > v) ? 1 : 0;
    float kept = (cnt < KSEL) ? v : 0.0f;

    red[t] = kept;
    __syncthreads();
    #pragma unroll
    for (int st = 128; st > 0; st >>= 1) {
        if (t < st) red[t] += red[t + st];
        __syncthreads();
    }
    g[(size_t)b * NB + t] = kept * ((float)NB / red[0]);
}

// ---------------------------------------------------------------------------
extern "C" void kernel_launch(void* const* d_in, const int* in_sizes, int n_in,
                              void* d_out, int out_size, void* d_ws, size_t ws_size,
                              hipStream_t stream) {
    const float* x      = (const float*)d_in[0];
    const float* weight = (const float*)d_in[1];
    const float* bias   = (const float*)d_in[2];
    const float* gate_w = (const float*)d_in[3];
    const float* gate_b = (const float*)d_in[4];
    float* out = (float*)d_out;
    float* g   = (float*)d_ws;               // BATCH*NB floats = 8 MiB scratch

    gate_gemm_kernel<<<dim3(BATCH / 16), 256, 0, stream>>>(x, gate_w, gate_b, g);
    sparsify_kernel<<<dim3(BATCH), 256, 0, stream>>>(g);
    gated_gemm_kernel<<<dim3(S, BATCH / 64), 256, 0, stream>>>(x, weight, bias, g, out);
}